// mse_weight_loss_86114094285238
// MI455X (gfx1250) — compile-verified
//
#include <hip/hip_runtime.h>

// Weighted-MSE reduction for [8,3,1080,1920] f32 tensors.
// Memory-bound: ~398 MB of reads @ 23.3 TB/s -> ~17 us floor.
// Strategy: float4 (b128) streaming loads, per-thread FMA accumulation,
// wave32 reduction via V_WMMA_F32_16X16X4_F32 (A = partial sums, B = ones),
// LDS block reduction, one global atomic per block, tiny finalize kernel.

typedef __attribute__((ext_vector_type(2))) float v2f;
typedef __attribute__((ext_vector_type(8))) float v8f;

#define PI_F 3.14159265358979323846f

// Fixed problem shape from the reference's setup_inputs().
constexpr int kH = 1080;
constexpr int kW = 1920;

__global__ void zero_ws_kernel(float* ws) { ws[0] = 0.0f; }

__global__ __launch_bounds__(256) void wloss_main_kernel(
    const float* __restrict__ x, const float* __restrict__ y,
    float* __restrict__ ws, unsigned n4) {
  const unsigned tid = blockIdx.x * blockDim.x + threadIdx.x;
  const unsigned stride = gridDim.x * blockDim.x;
  const float4* __restrict__ x4 = reinterpret_cast<const float4*>(x);
  const float4* __restrict__ y4 = reinterpret_cast<const float4*>(y);

  const float hscale = PI_F / (float)kH;
  const float hbias = 0.5f - 0.5f * (float)kH;

  float acc = 0.0f;
  for (unsigned g = tid; g < n4; g += stride) {
    float4 a = x4[g];          // global_load_b128
    float4 b = y4[g];          // global_load_b128
    unsigned idx = g * 4u;     // flat element index (W divisible by 4 ->
    unsigned h = (idx / (unsigned)kW) % (unsigned)kH;  // one row weight/grp)
    float wgt = __cosf(((float)h + hbias) * hscale);   // v_cos_f32
    float d0 = a.x - b.x;
    float d1 = a.y - b.y;
    float d2 = a.z - b.z;
    float d3 = a.w - b.w;
    float s = fmaf(d0, d0, fmaf(d1, d1, fmaf(d2, d2, d3 * d3)));
    acc = fmaf(wgt, s, acc);
  }

  // ---- wave32 reduction via WMMA ------------------------------------
  // A (16x4 f32, 2 VGPRs): VGPR0 = acc -> A[m,0]=acc[m], A[m,2]=acc[m+16];
  // VGPR1 = 0. B = all ones (4x16). D[m,n] = acc[m] + acc[m+16] (any n).
  // D VGPR r, lane L<16 holds D[r,L]; lane>=16 holds D[r+8,L-16].
  // Sum of the 8 D VGPRs: lanes<16 get sum over m=0..7 (+16 pairs),
  // lanes>=16 get m=8..15 (+16 pairs). shfl_xor(16) completes the sum.
  // EXEC is all 1s here: no divergent exits before this point.
  v2f A; A.x = acc; A.y = 0.0f;
  v2f B; B.x = 1.0f; B.y = 1.0f;
  v8f C = {};
  v8f D = __builtin_amdgcn_wmma_f32_16x16x4_f32(
      /*neg_a=*/false, A, /*neg_b=*/false, B,
      /*c_mod=*/(short)0, C, /*reuse_a=*/false, /*reuse_b=*/false);
  float t = ((D[0] + D[1]) + (D[2] + D[3])) + ((D[4] + D[5]) + (D[6] + D[7]));
  t += __shfl_xor(t, 16, 32);  // combine lane groups 0-15 and 16-31

  // ---- block reduction + one global atomic per block ----------------
  __shared__ float blk;
  if (threadIdx.x == 0) blk = 0.0f;
  __syncthreads();
  if ((threadIdx.x & 31u) == 0u) atomicAdd(&blk, t);  // ds_add_f32
  __syncthreads();
  if (threadIdx.x == 0) atomicAdd(ws, blk);           // global_atomic_add_f32
}

__global__ __launch_bounds__(256) void wloss_final_kernel(
    const float* __restrict__ ws, float* __restrict__ out) {
  __shared__ float red[8];
  const float hscale = PI_F / (float)kH;
  const float hbias = 0.5f - 0.5f * (float)kH;

  float s = 0.0f;
  for (int j = (int)threadIdx.x; j < kH; j += (int)blockDim.x)
    s += __cosf(((float)j + hbias) * hscale);

  for (int off = 16; off > 0; off >>= 1) s += __shfl_down(s, off, 32);
  if ((threadIdx.x & 31u) == 0u) red[threadIdx.x >> 5] = s;
  __syncthreads();
  if (threadIdx.x == 0) {
    float d = 0.0f;
#pragma unroll
    for (int i = 0; i < 8; ++i) d += red[i];
    out[0] = ws[0] / (d * (float)kW);  // denom = W * sum_h cos(...)
  }
}

extern "C" void kernel_launch(void* const* d_in, const int* in_sizes, int n_in,
                              void* d_out, int out_size, void* d_ws,
                              size_t ws_size, hipStream_t stream) {
  (void)n_in; (void)out_size; (void)ws_size;
  const float* x = (const float*)d_in[0];
  const float* y = (const float*)d_in[1];
  float* out = (float*)d_out;
  float* ws = (float*)d_ws;

  const unsigned n = (unsigned)in_sizes[0];  // 8*3*1080*1920 = 49,766,400
  const unsigned n4 = n / 4u;

  zero_ws_kernel<<<1, 1, 0, stream>>>(ws);

  // Enough blocks to saturate HBM; each thread streams ~8 float4 groups.
  unsigned blocks = (n4 + 255u) / 256u;
  if (blocks > 6144u) blocks = 6144u;
  wloss_main_kernel<<<blocks, 256, 0, stream>>>(x, y, ws, n4);

  wloss_final_kernel<<<1, 256, 0, stream>>>(ws, out);
}